// sta_16045997818332
// MI455X (gfx1250) — compile-verified
//
#include <hip/hip_runtime.h>

// EMA scan y_t = y_{t-1} + w*(x_t - y_{t-1}) over T for wave[4096][6000][3] f32.
// Memory-bound: 590 MB total @ 23.3 TB/s => ~25us floor. Formulated as blocked
// 16x16 matmuls (lower-triangular decay matrix L[i][j] = w*q^(i-j)) on the fp32
// WMMA path (V_WMMA_F32_16X16X4_F32, f32-in/f32-acc: the precision-correct
// matrix op for an fp32 reference). T is parallelized using decay truncation:
// q = 0.4, q^48 ~ 8e-20 << fp32 eps, so each T-chunk warms up 48 steps cold.

typedef __attribute__((ext_vector_type(2))) float v2f;
typedef __attribute__((ext_vector_type(8))) float v8f;

#define WAVE        32
#define SEQS        12288          // 4096 * 3 independent (b,c) sequences
#define SEQ_TILES   768            // SEQS / 16 columns per WMMA tile
#define T_LEN       6000
#define SUB         16             // timesteps per WMMA tile (M and K dim)
#define CHUNK_SUB   25             // sub-tiles per T-chunk (400 timesteps)
#define N_CHUNKS    15             // 375 sub-tiles / 25
#define WARM_SUB    3              // 48-step warmup
#define ROW_STRIDE  18000          // 6000 * 3 floats per batch row

__device__ __forceinline__ v8f ema_tile(const v2f afrag[4], const float* pin,
                                        const v8f& cc, float carry)
{
    // Load X fragments (B matrices, 4x16 per K-chunk). All offsets are
    // compile-time immediates off pin -> global_load_b32 with inst offsets.
    v2f b0, b1, b2, b3;
    b0.x = pin[0];  b0.y = pin[3];
    b1.x = pin[12]; b1.y = pin[15];
    b2.x = pin[24]; b2.y = pin[27];
    b3.x = pin[36]; b3.y = pin[39];

    // Accumulator starts as decayed carry term: C[i][j] = q^(i+1)*carry[j]
    v8f acc;
    #pragma unroll
    for (int v = 0; v < 8; ++v) acc[v] = cc[v] * carry;

    // Y = L*X + carry via 4 chained fp32 WMMAs (16x16x4)
    acc = __builtin_amdgcn_wmma_f32_16x16x4_f32(false, afrag[0], false, b0, (short)0, acc, false, false);
    acc = __builtin_amdgcn_wmma_f32_16x16x4_f32(false, afrag[1], false, b1, (short)0, acc, false, false);
    acc = __builtin_amdgcn_wmma_f32_16x16x4_f32(false, afrag[2], false, b2, (short)0, acc, false, false);
    acc = __builtin_amdgcn_wmma_f32_16x16x4_f32(false, afrag[3], false, b3, (short)0, acc, false, false);
    return acc;
}

__global__ __launch_bounds__(128)
void sta_ema_wmma_kernel(const float* __restrict__ in, float* __restrict__ out)
{
    const float W   = 0.6f;
    const float L2Q = -1.32192809f;        // log2(0.4)

    const int lane   = threadIdx.x & (WAVE - 1);
    const int waveId = blockIdx.x * (blockDim.x / WAVE) + (threadIdx.x / WAVE);

    const int seqTile = waveId % SEQ_TILES;
    const int chunk   = waveId / SEQ_TILES;

    // -------- WMMA fragment geometry (wave32, 32-bit operands) --------
    // A 16x4 : lanes 0-15 hold M=lane, K=kk*4+{0,1}; lanes 16-31 M=lane-16, K=kk*4+{2,3}
    // B 4x16 : lanes 0-15 hold N=lane, low K pair; lanes 16-31 high K pair
    // C/D    : VGPR v, lanes 0-15 -> (M=v, N=lane); lanes 16-31 -> (M=v+8, N=lane-16)
    const int n  = lane & 15;              // tile column (sequence)
    const int hi = lane >> 4;              // half-wave select
    const int kb = hi * 2;                 // K offset inside a 4-wide K chunk
    const int m0 = hi * 8;                 // first C/D row held by this lane

    // A = L, lower-triangular decay matrix: L[i][j] = (i>=j) ? W*q^(i-j) : 0
    v2f afrag[4];
    #pragma unroll
    for (int kk = 0; kk < 4; ++kk) {
        const int k0 = kk * 4 + kb;
        const int d0 = n - k0, d1 = n - k0 - 1;
        afrag[kk].x = (d0 >= 0) ? W * exp2f((float)d0 * L2Q) : 0.0f;
        afrag[kk].y = (d1 >= 0) ? W * exp2f((float)d1 * L2Q) : 0.0f;
    }

    // carry-injection coefficients: q^(m0+v+1)
    v8f cc;
    #pragma unroll
    for (int v = 0; v < 8; ++v)
        cc[v] = exp2f((float)(m0 + v + 1) * L2Q);

    // sequence -> memory mapping: s = 3*b + c; elem (s,t) at b*18000 + t*3 + c
    const int  s    = seqTile * 16 + n;
    const long base = (long)(s / 3) * ROW_STRIDE + (s % 3);

    const int warm = (chunk == 0) ? 0 : WARM_SUB;
    const int t0   = (chunk * CHUNK_SUB - warm) * SUB;

    // Running per-lane pointers: all in-loop accesses use immediate offsets.
    const float* pin  = in  + base + (long)t0 * 3 + kb * 3;
    float*       pout = out + base + (long)(t0 + warm * SUB) * 3 + m0 * 3;

    float carry = 0.0f;

    // ---- warmup: converge carry from cold start; no stores, no EXEC games ----
    for (int st = 0; st < warm; ++st) {
        v8f acc = ema_tile(afrag, pin, cc, carry);
        carry = __shfl(acc[7], 16 + n, WAVE);   // Y[15][n] lives in lane 16+n, VGPR7
        pin += SUB * 3;
    }

    // ---- main loop: load tile, 4x WMMA, broadcast carry, store 16 rows ----
    for (int st = 0; st < CHUNK_SUB; ++st) {
        __builtin_prefetch(pin + SUB * 3, 0, 1);      // global_prefetch_b8
        v8f acc = ema_tile(afrag, pin, cc, carry);
        carry = __shfl(acc[7], 16 + n, WAVE);

        #pragma unroll
        for (int v = 0; v < 8; ++v)
            pout[v * 3] = acc[v];                     // immediate-offset stores

        pin  += SUB * 3;
        pout += SUB * 3;
    }
}

extern "C" void kernel_launch(void* const* d_in, const int* in_sizes, int n_in,
                              void* d_out, int out_size, void* d_ws, size_t ws_size,
                              hipStream_t stream) {
    const float* wave = (const float*)d_in[0];
    float*       out  = (float*)d_out;
    (void)in_sizes; (void)n_in; (void)out_size; (void)d_ws; (void)ws_size;

    // 768 seq-tiles * 15 T-chunks = 11520 waves; 4 waves per 128-thread block
    const int totalWaves = SEQ_TILES * N_CHUNKS;
    const int blocks     = totalWaves / 4;   // 2880
    sta_ema_wmma_kernel<<<blocks, 128, 0, stream>>>(wave, out);
}